// GCN_31774168055916
// MI455X (gfx1250) — compile-verified
//
#include <hip/hip_runtime.h>

typedef __attribute__((ext_vector_type(2))) float v2f;
typedef __attribute__((ext_vector_type(8))) float v8f;

// ---------------------------------------------------------------------------
// Degree / normalization kernels
// ---------------------------------------------------------------------------
__global__ void init_deg_kernel(float* __restrict__ deg, int n) {
    int i = blockIdx.x * blockDim.x + threadIdx.x;
    if (i < n) deg[i] = 1.0f;  // self-loop contributes 1
}

__global__ void edge_deg_kernel(const int* __restrict__ dst, float* __restrict__ deg, int e) {
    int i = blockIdx.x * blockDim.x + threadIdx.x;
    if (i < e) atomicAdd(&deg[dst[i]], 1.0f);
}

__global__ void deg_to_dinv_kernel(float* __restrict__ d, int n) {
    int i = blockIdx.x * blockDim.x + threadIdx.x;
    if (i < n) d[i] = rsqrtf(d[i]);  // deg >= 1 always (self-loops)
}

// ---------------------------------------------------------------------------
// fp32 WMMA GEMM:  C[M,N] = A[M,K] @ B[K,N]
// One wave computes a 16 x (NTILES*16) slab of C using V_WMMA_F32_16X16X4_F32.
// Requires M % 16 == 0 and K % 4 == 0.  NGUARD handles N not multiple of 16.
//
// Fragment layouts (CDNA5 ISA 7.12.2, f32 16x16x4):
//   A (16x4):  lane -> M = lane%16,  vgpr v -> K = 2*(lane/16) + v
//   B (4x16):  lane -> N = lane%16,  vgpr v -> K = 2*(lane/16) + v
//   C (16x16): lane -> N = lane%16,  vgpr v -> M = v + 8*(lane/16)
// ---------------------------------------------------------------------------
template <int NTILES, bool NGUARD>
__global__ void gemm_wmma_kernel(const float* __restrict__ A,
                                 const float* __restrict__ B,
                                 float* __restrict__ C,
                                 int M, int N, int K) {
    const int lane  = threadIdx.x & 31;
    const int wave  = threadIdx.x >> 5;
    const int mtile = blockIdx.x * (blockDim.x >> 5) + wave;
    const int mtiles = M >> 4;
    if (mtile >= mtiles) return;  // uniform per-wave guard: EXEC stays all-1s

    const int half = lane >> 4;   // 0 or 1
    const int lm   = lane & 15;

    v8f acc[NTILES] = {};

    const float* arow = A + (size_t)(mtile * 16 + lm) * K;

    for (int k = 0; k < K; k += 4) {
        v2f a;
        a.x = arow[k + 2 * half + 0];
        a.y = arow[k + 2 * half + 1];
#pragma unroll
        for (int t = 0; t < NTILES; ++t) {
            const int col = t * 16 + lm;
            v2f b;
            if (!NGUARD || col < N) {
                b.x = B[(size_t)(k + 2 * half + 0) * N + col];
                b.y = B[(size_t)(k + 2 * half + 1) * N + col];
            } else {
                b.x = 0.0f; b.y = 0.0f;
            }
            acc[t] = __builtin_amdgcn_wmma_f32_16x16x4_f32(
                /*neg_a=*/false, a, /*neg_b=*/false, b,
                /*c_mod=*/(short)0, acc[t], /*reuse_a=*/false, /*reuse_b=*/false);
        }
    }

#pragma unroll
    for (int t = 0; t < NTILES; ++t) {
        const int col = t * 16 + lm;
        if (NGUARD && col >= N) continue;
#pragma unroll
        for (int v = 0; v < 8; ++v) {
            const int row = mtile * 16 + v + 8 * half;
            C[(size_t)row * N + col] = acc[t][v];
        }
    }
}

// ---------------------------------------------------------------------------
// out[i,f] = bias[f] + tmp[i,f] * dinv[i]^2     (self-loop term + bias;
// also fully initializes the poisoned output buffer before the edge scatter)
// ---------------------------------------------------------------------------
__global__ void init_out_kernel(const float* __restrict__ tmp,
                                const float* __restrict__ bias,
                                const float* __restrict__ dinv,
                                float* __restrict__ out,
                                int n, int F) {
    int i = blockIdx.x * blockDim.x + threadIdx.x;
    int total = n * F;
    if (i >= total) return;
    int node = i / F;
    int f    = i - node * F;
    float di = dinv[node];
    out[i] = bias[f] + tmp[i] * di * di;
}

// ---------------------------------------------------------------------------
// Edge scatter: out[dst] += tmp[src] * dinv[src]*dinv[dst]
// One thread handles 4 consecutive features of one edge (F % 4 == 0:
// 128/4=32, 40/4=10).  float4 gather is coalesced along the feature dim;
// scatter uses per-dword global_atomic_add_f32.
// ---------------------------------------------------------------------------
__global__ void scatter_edges_kernel(const int* __restrict__ src,
                                     const int* __restrict__ dst,
                                     const float* __restrict__ dinv,
                                     const float* __restrict__ tmp,
                                     float* __restrict__ out,
                                     int e, int F) {
    const int qper = F >> 2;
    int t = blockIdx.x * blockDim.x + threadIdx.x;
    int total = e * qper;
    if (t >= total) return;
    int edge = t / qper;
    int q    = (t - edge * qper) << 2;
    int s = src[edge];
    int d = dst[edge];
    float norm = dinv[s] * dinv[d];
    const float4 v = *(const float4*)(tmp + (size_t)s * F + q);
    float* op = out + (size_t)d * F + q;
    atomicAdd(op + 0, v.x * norm);
    atomicAdd(op + 1, v.y * norm);
    atomicAdd(op + 2, v.z * norm);
    atomicAdd(op + 3, v.w * norm);
}

// ---------------------------------------------------------------------------
// Launch
// ---------------------------------------------------------------------------
extern "C" void kernel_launch(void* const* d_in, const int* in_sizes, int n_in,
                              void* d_out, int out_size, void* d_ws, size_t ws_size,
                              hipStream_t stream) {
    const float* x  = (const float*)d_in[0];
    const int*   ei = (const int*)d_in[1];
    const float* W1 = (const float*)d_in[2];
    const float* b1 = (const float*)d_in[3];
    const float* W2 = (const float*)d_in[4];
    const float* b2 = (const float*)d_in[5];
    const float* W3 = (const float*)d_in[6];
    const float* b3 = (const float*)d_in[7];

    const int nH  = in_sizes[3];            // 128
    const int nC  = in_sizes[7];            // 40
    const int inF = in_sizes[2] / nH;       // 256
    const int n   = in_sizes[0] / inF;      // 50000
    const int e   = in_sizes[1] / 2;        // 800000

    const int* src = ei;
    const int* dst = ei + e;

    float* ws   = (float*)d_ws;
    size_t off0 = ((size_t)n + 255) & ~(size_t)255;   // dinv slot, padded
    float* dinv = ws;
    float* tmpb = ws + off0;                          // 50000 x 128 scratch (>= 50000x40)
    float* hbuf = tmpb + (size_t)n * nH;              // 50000 x 128 activations
    float* out  = (float*)d_out;                      // 50000 x 40

    const int TB = 256;

    // --- symmetric normalization -------------------------------------------
    init_deg_kernel<<<(n + TB - 1) / TB, TB, 0, stream>>>(dinv, n);
    edge_deg_kernel<<<(e + TB - 1) / TB, TB, 0, stream>>>(dst, dinv, e);
    deg_to_dinv_kernel<<<(n + TB - 1) / TB, TB, 0, stream>>>(dinv, n);

    const int mtiles  = n >> 4;                       // 3125 (n % 16 == 0)
    const int gblocks = (mtiles + 3) / 4;             // 4 waves (M-tiles) per block

    // --- layer 1: x[N,256] @ W1[256,128] -----------------------------------
    gemm_wmma_kernel<8, false><<<gblocks, 128, 0, stream>>>(x, W1, tmpb, n, nH, inF);
    init_out_kernel<<<((n * nH) + TB - 1) / TB, TB, 0, stream>>>(tmpb, b1, dinv, hbuf, n, nH);
    scatter_edges_kernel<<<((e * (nH >> 2)) + TB - 1) / TB, TB, 0, stream>>>(
        src, dst, dinv, tmpb, hbuf, e, nH);

    // --- layer 2: h1[N,128] @ W2[128,128] ----------------------------------
    gemm_wmma_kernel<8, false><<<gblocks, 128, 0, stream>>>(hbuf, W2, tmpb, n, nH, nH);
    init_out_kernel<<<((n * nH) + TB - 1) / TB, TB, 0, stream>>>(tmpb, b2, dinv, hbuf, n, nH);
    scatter_edges_kernel<<<((e * (nH >> 2)) + TB - 1) / TB, TB, 0, stream>>>(
        src, dst, dinv, tmpb, hbuf, e, nH);

    // --- layer 3: h2[N,128] @ W3[128,40] -----------------------------------
    gemm_wmma_kernel<3, true><<<gblocks, 128, 0, stream>>>(hbuf, W3, tmpb, n, nC, nH);
    init_out_kernel<<<((n * nC) + TB - 1) / TB, TB, 0, stream>>>(tmpb, b3, dinv, out, n, nC);
    scatter_edges_kernel<<<((e * (nC >> 2)) + TB - 1) / TB, TB, 0, stream>>>(
        src, dst, dinv, tmpb, out, e, nC);
}